// ConvBlock_5231270166892
// MI455X (gfx1250) — compile-verified
//
#include <hip/hip_runtime.h>

typedef float v2f __attribute__((ext_vector_type(2)));
typedef float v8f __attribute__((ext_vector_type(8)));

constexpr int   Bc    = 8;
constexpr int   Vc    = 12288;
constexpr int   Ec    = 98304;
constexpr int   FINc  = 256;
constexpr int   FOUTc = 256;
constexpr int   NROWS = Vc * Bc;            // 98304
constexpr float EPSc  = 1e-5f;

// ---------------------------------------------------------------- CSR build
__global__ void k_zero_i32(int* __restrict__ p, int n) {
    int i = blockIdx.x * 256 + threadIdx.x;
    if (i < n) p[i] = 0;
}

__global__ void k_degree(const int* __restrict__ dst, int* __restrict__ deg) {
    int e = blockIdx.x * 256 + threadIdx.x;
    if (e < Ec) atomicAdd(&deg[dst[e]], 1);
}

// one block, 256 threads: exclusive scan of deg[V] -> row_ptr[V+1], cursor[V]
__global__ void k_scan(const int* __restrict__ deg, int* __restrict__ row_ptr,
                       int* __restrict__ cursor) {
    __shared__ int tsum[256];
    __shared__ int tpre[256];
    const int chunk = Vc / 256;  // 48
    int t = threadIdx.x;
    int s = 0;
    for (int j = 0; j < chunk; ++j) s += deg[t * chunk + j];
    tsum[t] = s;
    __syncthreads();
    if (t == 0) {
        int acc = 0;
        for (int j = 0; j < 256; ++j) { tpre[j] = acc; acc += tsum[j]; }
    }
    __syncthreads();
    int acc = tpre[t];
    for (int j = 0; j < chunk; ++j) {
        int idx = t * chunk + j;
        row_ptr[idx] = acc;
        cursor[idx]  = acc;
        acc += deg[idx];
    }
    if (t == 0) row_ptr[Vc] = Ec;
}

__global__ void k_fill(const int* __restrict__ src, const int* __restrict__ dst,
                       const float* __restrict__ w, int* __restrict__ cursor,
                       int* __restrict__ col, float* __restrict__ wc) {
    int e = blockIdx.x * 256 + threadIdx.x;
    if (e >= Ec) return;
    int pos = atomicAdd(&cursor[dst[e]], 1);
    col[pos] = src[e];
    wc[pos]  = w[e];
}

// ---------------------------------------------------------------- L apply (gather)
// out[v*2048 + j] = alpha * sum_e w_e * in[src*in_sv + b*in_sb + i]  (- prev)
__global__ void k_lapply(float* __restrict__ out, const float* __restrict__ in,
                         const float* __restrict__ prev, const float* __restrict__ wc,
                         const int* __restrict__ col, const int* __restrict__ rp,
                         float alpha, int in_sv, int in_sb, int pv, int pb) {
    int v   = blockIdx.x;
    int beg = rp[v], end = rp[v + 1];
    int t   = threadIdx.x;
    #pragma unroll
    for (int s2 = 0; s2 < 8; ++s2) {
        int j = t + s2 * 256;            // 0..2047  (j = b*FIN + i)
        int b = j >> 8, i = j & 255;
        float acc = 0.f;
        for (int e = beg; e < end; ++e) {
            int   s = col[e];
            acc += wc[e] * in[(size_t)s * in_sv + (size_t)b * in_sb + i];
        }
        float r = alpha * acc;
        if (prev) r -= prev[(size_t)v * pv + (size_t)b * pb + i];
        out[(size_t)v * 2048 + j] = r;
    }
}

// ---------------------------------------------------------------- WMMA GEMM
// rows r in [0, V*B): v = r>>3, b = r&7 ; A-source: T_k (row-contig) or x (transposed idx)
// out_pre[(b*V+v)*FOUT + o] = sum_k sum_i Tk[r][i] * W[k][i][o] + bias[o]
__global__ __launch_bounds__(256) void k_gemm(
    const float* __restrict__ x,  const float* __restrict__ t1,
    const float* __restrict__ t2, const float* __restrict__ t3,
    const float* __restrict__ W,  const float* __restrict__ bias,
    float* __restrict__ out) {
    const int tid  = threadIdx.x;
    const int wave = tid >> 5;
    const int ln   = tid & 31;
    const int half = ln >> 4;          // K-pair selector within fragment
    const int l16  = ln & 15;
    const int wm   = wave & 1;         // 2 waves along M
    const int wn   = wave >> 1;        // 4 waves along N
    const int row0 = blockIdx.x * 128 + wm * 64;
    const int col0 = blockIdx.y * 128 + wn * 32;

    v8f acc[4][2];
    #pragma unroll
    for (int m = 0; m < 4; ++m)
        #pragma unroll
        for (int n = 0; n < 2; ++n)
            acc[m][n] = (v8f)0.f;

    const float* srcs[4] = {x, t1, t2, t3};

    for (int k = 0; k < 4; ++k) {
        const float* src = srcs[k];
        const float* ap[4];
        #pragma unroll
        for (int m = 0; m < 4; ++m) {
            int r = row0 + m * 16 + l16;
            size_t base = (k == 0)
                ? ((size_t)(r & 7) * Vc + (size_t)(r >> 3)) * FINc  // x: [B,V,FIN]
                : (size_t)r * FINc;                                 // Tk: [V*B,FIN]
            ap[m] = src + base;
        }
        const float* wk = W + (size_t)k * FINc * FOUTc;

        #pragma unroll 4
        for (int i = 0; i < FINc; i += 4) {
            const int brow = i + 2 * half;
            v2f bf[2];
            #pragma unroll
            for (int n = 0; n < 2; ++n) {
                int o = col0 + n * 16 + l16;
                bf[n].x = wk[(size_t)brow * FOUTc + o];
                bf[n].y = wk[(size_t)(brow + 1) * FOUTc + o];
            }
            v2f af[4];
            #pragma unroll
            for (int m = 0; m < 4; ++m) {
                const float* p = ap[m] + i + 2 * half;
                af[m].x = p[0];
                af[m].y = p[1];
            }
            #pragma unroll
            for (int m = 0; m < 4; ++m)
                #pragma unroll
                for (int n = 0; n < 2; ++n)
                    acc[m][n] = __builtin_amdgcn_wmma_f32_16x16x4_f32(
                        false, af[m], false, bf[n], (short)0, acc[m][n], false, false);
        }
    }

    // C/D layout: VGPR j -> M = j + 8*half, N = l16
    #pragma unroll
    for (int n = 0; n < 2; ++n) {
        int o = col0 + n * 16 + l16;
        float bo = bias[o];
        #pragma unroll
        for (int m = 0; m < 4; ++m) {
            #pragma unroll
            for (int j = 0; j < 8; ++j) {
                int r = row0 + m * 16 + j + 8 * half;
                size_t orow = ((size_t)(r & 7) * Vc + (size_t)(r >> 3)) * FOUTc;
                out[orow + o] = acc[m][n][j] + bo;
            }
        }
    }
}

// ---------------------------------------------------------------- BN stats (deterministic)
__global__ void k_stats_partial(const float* __restrict__ out, float* __restrict__ part) {
    const int c  = threadIdx.x;
    const int rb = NROWS / 192;          // 512 rows per block
    const int r0 = blockIdx.x * rb;
    float s = 0.f, q = 0.f;
    for (int r = 0; r < rb; ++r) {
        float v = out[(size_t)(r0 + r) * FOUTc + c];
        s += v;
        q += v * v;
    }
    part[(size_t)blockIdx.x * 512 + c]       = s;
    part[(size_t)blockIdx.x * 512 + 256 + c] = q;
}

__global__ void k_stats_reduce(const float* __restrict__ part, float* __restrict__ stats) {
    int c = threadIdx.x;
    float s = 0.f, q = 0.f;
    for (int bl = 0; bl < 192; ++bl) {
        s += part[bl * 512 + c];
        q += part[bl * 512 + 256 + c];
    }
    stats[c]       = s;
    stats[256 + c] = q;
}

__global__ void k_bnrelu(float* __restrict__ out, const float* __restrict__ stats,
                         const float* __restrict__ gamma, const float* __restrict__ beta) {
    #pragma unroll
    for (int s2 = 0; s2 < 4; ++s2) {
        int i = blockIdx.x * 1024 + s2 * 256 + threadIdx.x;
        int c = i & 255;
        float mean = stats[c] * (1.f / NROWS);
        float var  = stats[256 + c] * (1.f / NROWS) - mean * mean;
        float sc   = gamma[c] * rsqrtf(var + EPSc);
        float sh   = beta[c] - mean * sc;
        float y    = fmaf(out[i], sc, sh);
        out[i]     = y > 0.f ? y : 0.f;
    }
}

// ---------------------------------------------------------------- launcher
extern "C" void kernel_launch(void* const* d_in, const int* in_sizes, int n_in,
                              void* d_out, int out_size, void* d_ws, size_t ws_size,
                              hipStream_t stream) {
    (void)in_sizes; (void)n_in; (void)out_size; (void)ws_size;
    const float* x   = (const float*)d_in[0];
    const float* ew  = (const float*)d_in[1];
    const float* W   = (const float*)d_in[2];
    const float* bia = (const float*)d_in[3];
    const float* gam = (const float*)d_in[4];
    const float* bet = (const float*)d_in[5];
    const int*   esr = (const int*)d_in[6];
    const int*   eds = (const int*)d_in[7];
    float*       out = (float*)d_out;

    const size_t S = (size_t)Vc * Bc * FINc;   // 25,165,824 floats per T buffer
    size_t off = 0;
    auto alloc = [&](size_t bytes) {
        size_t o = off;
        off = (off + bytes + 255) & ~(size_t)255;
        return o;
    };
    char* ws = (char*)d_ws;
    float* t1      = (float*)(ws + alloc(S * 4));
    float* t2      = (float*)(ws + alloc(S * 4));
    float* t3      = (float*)(ws + alloc(S * 4));
    int*   deg     = (int*)  (ws + alloc((size_t)Vc * 4));
    int*   row_ptr = (int*)  (ws + alloc((size_t)(Vc + 1) * 4));
    int*   cursor  = (int*)  (ws + alloc((size_t)Vc * 4));
    int*   col     = (int*)  (ws + alloc((size_t)Ec * 4));
    float* wcsr    = (float*)(ws + alloc((size_t)Ec * 4));
    float* part    = (float*)(ws + alloc((size_t)192 * 512 * 4));
    float* stats   = (float*)(ws + alloc((size_t)512 * 4));

    // CSR build
    k_zero_i32<<<(Vc + 255) / 256, 256, 0, stream>>>(deg, Vc);
    k_degree  <<<Ec / 256, 256, 0, stream>>>(eds, deg);
    k_scan    <<<1, 256, 0, stream>>>(deg, row_ptr, cursor);
    k_fill    <<<Ec / 256, 256, 0, stream>>>(esr, eds, ew, cursor, col, wcsr);

    // Chebyshev recursion (gather formulation)
    // T1 = L x              (in = x: node stride FIN, batch stride V*FIN)
    k_lapply<<<Vc, 256, 0, stream>>>(t1, x, nullptr, wcsr, col, row_ptr,
                                     1.f, FINc, Vc * FINc, 0, 0);
    // T2 = 2 L T1 - x
    k_lapply<<<Vc, 256, 0, stream>>>(t2, t1, x, wcsr, col, row_ptr,
                                     2.f, Bc * FINc, FINc, FINc, Vc * FINc);
    // T3 = 2 L T2 - T1
    k_lapply<<<Vc, 256, 0, stream>>>(t3, t2, t1, wcsr, col, row_ptr,
                                     2.f, Bc * FINc, FINc, Bc * FINc, FINc);

    // WMMA GEMM: out_pre = sum_k Tk @ W[k] + bias
    k_gemm<<<dim3(NROWS / 128, FOUTc / 128), 256, 0, stream>>>(x, t1, t2, t3, W, bia, out);

    // BatchNorm (training stats) + ReLU
    k_stats_partial<<<192, 256, 0, stream>>>(out, part);
    k_stats_reduce <<<1, 256, 0, stream>>>(part, stats);
    k_bnrelu<<<(int)((size_t)NROWS * FOUTc / 1024), 256, 0, stream>>>(out, stats, gam, bet);
}